// Mutation_GCN_12232066859616
// MI455X (gfx1250) — compile-verified
//
#include <hip/hip_runtime.h>
#include <math.h>

#define NNODES 100000
#define NEDGES 1600000
#define FIN    128
#define HID    64
#define NCLS   40
#define NCLSP  48   // padded to 3 tiles of 16

typedef __attribute__((ext_vector_type(2))) float v2f;
typedef __attribute__((ext_vector_type(8))) float v8f;

__global__ void zero_kernel(float* __restrict__ p, long n) {
    long i = (long)blockIdx.x * blockDim.x + threadIdx.x;
    long stride = (long)gridDim.x * blockDim.x;
    for (; i < n; i += stride) p[i] = 0.0f;
}

__global__ void degree_kernel(const int* __restrict__ dst, float* __restrict__ deg, int e) {
    int i = blockIdx.x * blockDim.x + threadIdx.x;
    if (i < e) atomicAdd(&deg[dst[i]], 1.0f);
}

// deg -> dinv in place; +1 accounts for the self-loop (deg >= 1 always)
__global__ void dinv_kernel(float* __restrict__ deg, int n) {
    int i = blockIdx.x * blockDim.x + threadIdx.x;
    if (i < n) deg[i] = rsqrtf(deg[i] + 1.0f);
}

__global__ void pad_w2_kernel(const float* __restrict__ W2, float* __restrict__ W2p) {
    int i = blockIdx.x * blockDim.x + threadIdx.x;
    if (i >= HID * NCLSP) return;
    int r = i / NCLSP, c = i - r * NCLSP;
    W2p[i] = (c < NCLS) ? W2[r * NCLS + c] : 0.0f;
}

// C[M x (NT*16)] = A[M x K] @ B[K x (NT*16)], fp32 WMMA 16x16x4, LDB == NT*16.
// B is staged in LDS in K-pair-packed layout: pair p = k/2 holds
// {B[2p][c], B[2p+1][c]} contiguously, so each lane's B operand is one
// 8-byte-aligned ds_load_b64 straight into an even VGPR pair.
// Each wave computes one 16-row strip across all NT column tiles: every A
// element is loaded exactly once (global_load_b64) and feeds NT WMMAs.
template <int NT>
__global__ void gemm_wmma_lds(const float* __restrict__ A, int lda,
                              const float* __restrict__ B,
                              float* __restrict__ C, int ldc,
                              int ntm, int K) {
    constexpr int LDB = NT * 16;
    extern __shared__ float sB[];
    // cooperative stage of B into LDS, packing K-pairs contiguously
    for (int i = threadIdx.x; i < K * LDB; i += blockDim.x) {
        int k = i / LDB;
        int c = i - k * LDB;
        sB[(k >> 1) * (2 * LDB) + 2 * c + (k & 1)] = B[i];
    }
    __syncthreads();

    int wave = blockIdx.x * (blockDim.x >> 5) + (threadIdx.x >> 5);
    if (wave >= ntm) return;                 // wave-uniform: EXEC stays all-ones
    int lane = threadIdx.x & 31;
    int half = lane >> 4;                    // 0: K pair {k,k+1}; 1: {k+2,k+3}
    int l    = lane & 15;

    const float* Arow = A + (long)(wave * 16 + l) * lda + 2 * half;
    const v2f*   sB2  = (const v2f*)sB + half * LDB + l;   // pair-indexed view

    v8f acc[NT];
    #pragma unroll
    for (int t = 0; t < NT; ++t) acc[t] = {};

    #pragma unroll 2
    for (int k = 0; k < K; k += 4) {
        v2f a;
        a.x = Arow[k];
        a.y = Arow[k + 1];
        const v2f* bp = sB2 + (k >> 1) * LDB;
        #pragma unroll
        for (int t = 0; t < NT; ++t) {
            v2f b = bp[t * 16];
            acc[t] = __builtin_amdgcn_wmma_f32_16x16x4_f32(
                false, a, false, b, (short)0, acc[t], false, false);
        }
    }

    float* Crow = C + (long)(wave * 16 + 8 * half) * ldc + l;
    #pragma unroll
    for (int t = 0; t < NT; ++t)
        #pragma unroll
        for (int r = 0; r < 8; ++r)
            Crow[(long)r * ldc + t * 16] = acc[t][r];
}

// acc[dst] += dinv[src]*dinv[dst] * feat[src]; one thread per edge, float4 gathers.
template <int F, int LDF, int LDA>
__global__ void agg_edges(const int* __restrict__ src, const int* __restrict__ dst,
                          const float* __restrict__ dinv,
                          const float* __restrict__ feat,
                          float* __restrict__ acc, int e) {
    int i = blockIdx.x * blockDim.x + threadIdx.x;
    if (i >= e) return;
    int s = src[i], d = dst[i];
    float nrm = dinv[s] * dinv[d];
    const float4* fp = (const float4*)(feat + (long)s * LDF);
    float* ap = acc + (long)d * LDA;
    #pragma unroll 4
    for (int c = 0; c < F / 4; ++c) {
        float4 v = fp[c];
        atomicAdd(ap + 4 * c + 0, nrm * v.x);
        atomicAdd(ap + 4 * c + 1, nrm * v.y);
        atomicAdd(ap + 4 * c + 2, nrm * v.z);
        atomicAdd(ap + 4 * c + 3, nrm * v.w);
    }
}

// acc = relu(acc + dinv^2 * feat + bias)   (layer-1 epilogue; adds self-loop msg)
__global__ void finish_relu(float* __restrict__ acc, const float* __restrict__ feat,
                            const float* __restrict__ dinv, const float* __restrict__ bias,
                            int n) {
    int idx = blockIdx.x * blockDim.x + threadIdx.x;
    if (idx >= n) return;
    int i = idx >> 6;            // HID == 64
    int f = idx & 63;
    float di = dinv[i];
    float v = acc[idx] + di * di * feat[idx] + bias[f];
    acc[idx] = v > 0.0f ? v : 0.0f;
}

// logits = acc2 + dinv^2 * lt + b2; out = log_softmax(logits)
__global__ void finish_logsoftmax(const float* __restrict__ acc2, const float* __restrict__ lt,
                                  const float* __restrict__ dinv, const float* __restrict__ b2,
                                  float* __restrict__ out, int n) {
    int i = blockIdx.x * blockDim.x + threadIdx.x;
    if (i >= n) return;
    float di = dinv[i];
    float di2 = di * di;
    const float* a = acc2 + (long)i * NCLS;
    const float* t = lt   + (long)i * NCLSP;
    float row[NCLS];
    float mx = -3.402823466e38f;
    #pragma unroll
    for (int c = 0; c < NCLS; ++c) {
        row[c] = a[c] + di2 * t[c] + b2[c];
        mx = fmaxf(mx, row[c]);
    }
    float s = 0.0f;
    #pragma unroll
    for (int c = 0; c < NCLS; ++c) s += expf(row[c] - mx);
    float lse = mx + logf(s);
    float* o = out + (long)i * NCLS;
    #pragma unroll
    for (int c = 0; c < NCLS; ++c) o[c] = row[c] - lse;
}

extern "C" void kernel_launch(void* const* d_in, const int* in_sizes, int n_in,
                              void* d_out, int out_size, void* d_ws, size_t ws_size,
                              hipStream_t stream) {
    const float* x  = (const float*)d_in[0];   // [N, 128]
    const int*   ei = (const int*)  d_in[1];   // [2, E]
    const float* W1 = (const float*)d_in[2];   // [128, 64]
    const float* b1 = (const float*)d_in[3];   // [64]
    const float* W2 = (const float*)d_in[4];   // [64, 40]
    const float* b2 = (const float*)d_in[5];   // [40]
    const int* srcv = ei;
    const int* dstv = ei + NEDGES;

    float* ws   = (float*)d_ws;
    float* deg  = ws;                               // N         (becomes dinv)
    float* acc1 = deg  + NNODES;                    // N*64
    float* acc2 = acc1 + (long)NNODES * HID;        // N*40
    float* h1t  = acc2 + (long)NNODES * NCLS;       // N*64
    float* lt   = h1t  + (long)NNODES * HID;        // N*48
    float* W2p  = lt   + (long)NNODES * NCLSP;      // 64*48 (offset is 16B-aligned)

    // 1) zero contiguous accumulator region: deg | acc1 | acc2
    long zn = (long)NNODES * (1 + HID + NCLS);
    zero_kernel<<<2048, 256, 0, stream>>>(deg, zn);

    // 2) degrees + 3) dinv
    degree_kernel<<<(NEDGES + 255) / 256, 256, 0, stream>>>(dstv, deg, NEDGES);
    dinv_kernel<<<(NNODES + 255) / 256, 256, 0, stream>>>(deg, NNODES);

    // 4) pad W2 -> [64, 48]
    pad_w2_kernel<<<(HID * NCLSP + 255) / 256, 256, 0, stream>>>(W2, W2p);

    int ntm = NNODES / 16;                 // 6250 row strips, 8 waves/block
    int blocks = (ntm + 7) / 8;

    // 5) h1t = x @ W1  (WMMA fp32, W1 in LDS: 128*64*4 = 32 KB)
    gemm_wmma_lds<HID / 16><<<blocks, 256, FIN * HID * 4, stream>>>(
        x, FIN, W1, h1t, HID, ntm, FIN);

    // 6) scatter layer 1
    agg_edges<HID, HID, HID><<<(NEDGES + 255) / 256, 256, 0, stream>>>(
        srcv, dstv, deg, h1t, acc1, NEDGES);

    // 7) epilogue: self-loop + bias + ReLU (in place; acc1 becomes h)
    finish_relu<<<(NNODES * HID + 255) / 256, 256, 0, stream>>>(acc1, h1t, deg, b1,
                                                                NNODES * HID);

    // 8) lt = h @ W2p  (WMMA fp32, W2p in LDS: 64*48*4 = 12 KB)
    gemm_wmma_lds<NCLSP / 16><<<blocks, 256, HID * NCLSP * 4, stream>>>(
        acc1, HID, W2p, lt, NCLSP, ntm, HID);

    // 9) scatter layer 2 (only the 40 real classes)
    agg_edges<NCLS, NCLSP, NCLS><<<(NEDGES + 255) / 256, 256, 0, stream>>>(
        srcv, dstv, deg, lt, acc2, NEDGES);

    // 10) self-loop + bias + log_softmax -> d_out
    finish_logsoftmax<<<(NNODES + 255) / 256, 256, 0, stream>>>(acc2, lt, deg, b2,
                                                                (float*)d_out, NNODES);
}